// CompressedAttention_81372450390133
// MI455X (gfx1250) — compile-verified
//
#include <hip/hip_runtime.h>

// Problem constants (from reference)
#define B_   4
#define S_   4096
#define H_   2
#define D_   2
#define MD_  5

typedef __attribute__((ext_vector_type(2))) float v2f;
typedef __attribute__((ext_vector_type(8))) float v8f;

// ---------------------------------------------------------------------------
// Kernel 1: per-token preprocessing.
// s_qk = x . shared_qk_v ; s_v = x . shared_v_v
// Q = rope(rmsnorm(s_qk * q_u)) * (1/sqrt(D))   (1/sqrt(D) folded into Q)
// KV (interleaved, 16B/key) = { rope(rmsnorm(s_qk*k_u)), s_v*shared_vproj_u }
// ---------------------------------------------------------------------------
__global__ __launch_bounds__(256) void prep_kernel(
    const float* __restrict__ x, const float* __restrict__ freqs,
    const float* __restrict__ qk_v, const float* __restrict__ v_v,
    const float* __restrict__ vproj_u, const float* __restrict__ q_u,
    const float* __restrict__ k_u,
    float* __restrict__ Q, float* __restrict__ KV)
{
    int idx = blockIdx.x * blockDim.x + threadIdx.x;
    if (idx >= B_ * S_) return;
    int b = idx >> 12;            // / S_
    int s = idx & (S_ - 1);

    const float* xp = x + idx * MD_;
    float sqk = 0.f, sv = 0.f;
#pragma unroll
    for (int i = 0; i < MD_; ++i) {
        float xi = xp[i];
        sqk = fmaf(xi, qk_v[i], sqk);
        sv  = fmaf(xi, v_v[i],  sv);
    }

    float f  = freqs[s];          // D/2 == 1 -> single rotation angle
    float cs = __cosf(f);
    float sn = __sinf(f);
    const float inv_sqrt_d = 0.70710678118654752f;   // 1/sqrt(D), D=2

#pragma unroll
    for (int h = 0; h < H_; ++h) {
        float q0 = sqk * q_u[2 * h + 0];
        float q1 = sqk * q_u[2 * h + 1];
        float var = 0.5f * (q0 * q0 + q1 * q1);
        float r = 16.0f * rsqrtf(var + 1e-6f);
        q0 *= r; q1 *= r;
        float qr = q0 * cs - q1 * sn;
        float qi = q0 * sn + q1 * cs;
        int o = ((b * H_ + h) * S_ + s) * 2;
        Q[o + 0] = qr * inv_sqrt_d;
        Q[o + 1] = qi * inv_sqrt_d;
    }
    {
        float k0 = sqk * k_u[0];
        float k1 = sqk * k_u[1];
        float var = 0.5f * (k0 * k0 + k1 * k1);
        float r = 16.0f * rsqrtf(var + 1e-6f);
        k0 *= r; k1 *= r;
        int o = (b * S_ + s) * 4;
        KV[o + 0] = k0 * cs - k1 * sn;
        KV[o + 1] = k0 * sn + k1 * cs;
        KV[o + 2] = sv * vproj_u[0];
        KV[o + 3] = sv * vproj_u[1];
    }
}

// ---------------------------------------------------------------------------
// Kernel 2: attention. Block = 8 waves, each wave owns one 16-query tile; all
// waves in a block share batch b, whose full interleaved KV image (64 KB) is
// staged once into LDS (vectorized cooperative copy: 16 x b128 per thread).
// Score tile (16x16) = Q(16x4, D padded) x K^T(4x16) via v_wmma_f32_16x16x4_f32.
// Two passes over LDS: (1) row max, (2) exp-sum + P*V; shuffle reductions only
// outside the key loops.
// ---------------------------------------------------------------------------
__global__ __launch_bounds__(256) void attn_kernel(
    const float* __restrict__ Q, const float* __restrict__ KV,
    float* __restrict__ O)
{
    __shared__ float smem[S_ * 4];            // 64 KB: {k0,k1,v0,v1} per key

    const int tile0 = blockIdx.x * 8;         // first tile of this block
    const int b_blk = tile0 >> 9;             // tile>>8 = b*H+h ; >>1 = b
    const float* KVb = KV + (size_t)b_blk * S_ * 4;

    {   // cooperative staging: 4096 float4s across 256 threads (b128 traffic)
        const float4* src = (const float4*)KVb;
        float4* dst = (float4*)smem;
#pragma unroll 4
        for (int i = threadIdx.x; i < S_; i += 256) dst[i] = src[i];
    }
    __syncthreads();

    const int lane = threadIdx.x & 31;
    const int wv   = threadIdx.x >> 5;
    const int tile = tile0 + wv;               // 0 .. B*H*(S/16)-1
    const int qt   = tile & ((S_ / 16) - 1);   // query tile within sequence
    const int bh   = tile >> 8;                // b*H + h
    const int m    = lane & 15;
    const int half = lane >> 4;                // lanes 16-31 carry K=2,3 (pad)

    // A matrix (16x4 f32): row = m, lanes 0-15 hold K=0,1; lanes 16-31 pad=0
    const float* qp = Q + ((size_t)bh * S_ + (size_t)qt * 16 + m) * 2;
    float q0 = qp[0], q1 = qp[1];              // all lanes load (no divergence)
    v2f amat;
    amat.x = half ? 0.f : q0;                  // v_cndmask
    amat.y = half ? 0.f : q1;

    const float2* kv2 = (const float2*)smem;   // [key*2 + 0] = (k0,k1)
    const float4* kv4 = (const float4*)smem;   // [key]       = (k0,k1,v0,v1)

    // ---------------- pass 1: per-row max ----------------
    float tmax[8];
#pragma unroll
    for (int r = 0; r < 8; ++r) tmax[r] = -3.0e38f;

    for (int kt = 0; kt < S_ / 16; ++kt) {
        float2 kk = kv2[(kt * 16 + m) * 2];    // ds_load_b64 (broadcast hi half)
        v2f bmat;
        bmat.x = half ? 0.f : kk.x;            // v_cndmask, zero the K-pad lanes
        bmat.y = half ? 0.f : kk.y;
        v8f c = {};
        c = __builtin_amdgcn_wmma_f32_16x16x4_f32(
                false, amat, false, bmat, (short)0, c, false, false);
#pragma unroll
        for (int r = 0; r < 8; ++r) tmax[r] = fmaxf(tmax[r], c[r]);
    }

    float rowmax[8];
#pragma unroll
    for (int r = 0; r < 8; ++r) {
        float v = tmax[r];
#pragma unroll
        for (int mask = 1; mask < 16; mask <<= 1)
            v = fmaxf(v, __shfl_xor(v, mask, 32));
        rowmax[r] = v;
    }

    // ---------------- pass 2: exp-sum + P*V ----------------
    float lsum[8], a0[8], a1[8];
#pragma unroll
    for (int r = 0; r < 8; ++r) { lsum[r] = 0.f; a0[r] = 0.f; a1[r] = 0.f; }

    for (int kt = 0; kt < S_ / 16; ++kt) {
        float4 kk = kv4[kt * 16 + m];          // ds_load_b128: k0,k1,v0,v1
        v2f bmat;
        bmat.x = half ? 0.f : kk.x;
        bmat.y = half ? 0.f : kk.y;
        float v0 = kk.z;                       // V used by ALL lanes (column m)
        float v1 = kk.w;
        v8f c = {};
        c = __builtin_amdgcn_wmma_f32_16x16x4_f32(
                false, amat, false, bmat, (short)0, c, false, false);
#pragma unroll
        for (int r = 0; r < 8; ++r) {
            float p = __expf(c[r] - rowmax[r]);   // <= 1, no overflow
            lsum[r] += p;
            a0[r] = fmaf(p, v0, a0[r]);
            a1[r] = fmaf(p, v1, a1[r]);
        }
    }

    // final 16-lane reductions + store (row M = r + 8*half)
#pragma unroll
    for (int r = 0; r < 8; ++r) {
        float l  = lsum[r];
        float x0 = a0[r];
        float x1 = a1[r];
#pragma unroll
        for (int mask = 1; mask < 16; mask <<= 1) {
            l  += __shfl_xor(l,  mask, 32);
            x0 += __shfl_xor(x0, mask, 32);
            x1 += __shfl_xor(x1, mask, 32);
        }
        if (m == 0) {
            int row = qt * 16 + r + 8 * half;
            size_t o = ((size_t)bh * S_ + row) * 2;
            float inv = 1.0f / l;
            O[o + 0] = x0 * inv;
            O[o + 1] = x1 * inv;
        }
    }
}

// ---------------------------------------------------------------------------
// Kernel 3: collapse heads: s_o = out . shared_o_v ; final = s_o * o_u
// ---------------------------------------------------------------------------
__global__ __launch_bounds__(256) void combine_kernel(
    const float* __restrict__ O, const float* __restrict__ o_v,
    const float* __restrict__ o_u, float* __restrict__ out)
{
    int idx = blockIdx.x * blockDim.x + threadIdx.x;
    if (idx >= B_ * S_) return;
    int b = idx >> 12;
    int s = idx & (S_ - 1);
    size_t p0 = ((size_t)(b * H_ + 0) * S_ + s) * 2;
    size_t p1 = ((size_t)(b * H_ + 1) * S_ + s) * 2;
    float so = O[p0]     * o_v[0] + O[p0 + 1] * o_v[1]
             + O[p1]     * o_v[2] + O[p1 + 1] * o_v[3];
#pragma unroll
    for (int md = 0; md < MD_; ++md)
        out[(size_t)idx * MD_ + md] = so * o_u[md];
}

// ---------------------------------------------------------------------------
extern "C" void kernel_launch(void* const* d_in, const int* in_sizes, int n_in,
                              void* d_out, int out_size, void* d_ws, size_t ws_size,
                              hipStream_t stream)
{
    const float* x     = (const float*)d_in[0];
    const float* freqs = (const float*)d_in[1];
    const float* qk_v  = (const float*)d_in[2];
    const float* v_v   = (const float*)d_in[3];
    const float* o_v   = (const float*)d_in[4];
    const float* vproj = (const float*)d_in[5];
    const float* q_u   = (const float*)d_in[6];
    const float* k_u   = (const float*)d_in[7];
    const float* o_u   = (const float*)d_in[8];

    float* ws = (float*)d_ws;
    float* Q  = ws;                            // B*H*S*2 = 65536 floats
    float* KV = Q + (size_t)B_ * H_ * S_ * 2;  // B*S*4   = 65536 floats
    float* O  = KV + (size_t)B_ * S_ * 4;      // B*H*S*2 = 65536 floats
    float* out = (float*)d_out;

    prep_kernel<<<(B_ * S_ + 255) / 256, 256, 0, stream>>>(
        x, freqs, qk_v, v_v, vproj, q_u, k_u, Q, KV);

    attn_kernel<<<(B_ * H_ * (S_ / 16)) / 8, 256, 0, stream>>>(Q, KV, O);

    combine_kernel<<<(B_ * S_ + 255) / 256, 256, 0, stream>>>(O, o_v, o_u, out);
}